// Sampler_45913200394825
// MI455X (gfx1250) — compile-verified
//
#include <hip/hip_runtime.h>
#include <hip/hip_bf16.h>

typedef float v2f __attribute__((ext_vector_type(2)));
typedef float v4f __attribute__((ext_vector_type(4)));
typedef float v8f __attribute__((ext_vector_type(8)));

// ---------------------------------------------------------------------------
// Kernel 1: strided-pool reduction.
// pooled[b,c] = sum over even h, even w of b[b,c,h,w] * (1/65536)
// One 256-thread block per (b,c) plane (1024 planes). Each plane: 128 even
// rows * 64 float4 = 8192 float4 items -> 32 per thread, fully coalesced,
// nontemporal (stream-once data; don't thrash L2).
// ---------------------------------------------------------------------------
__global__ __launch_bounds__(256) void pool_even_kernel(
    const float* __restrict__ bsrc,   // [8,128,256,256]
    float* __restrict__ pooled)       // [1024] = [8,128]
{
    const int plane = blockIdx.x;              // 0..1023
    const int tid   = threadIdx.x;             // 0..255
    const float* base = bsrc + (size_t)plane * 65536u;

    float sum = 0.0f;
#pragma unroll
    for (int i = 0; i < 32; ++i) {
        const int item = tid + 256 * i;        // 0..8191
        const int row  = item >> 6;            // 0..127  (h = 2*row)
        const int qw   = item & 63;            // float4 index within row
        const v4f* p4 = reinterpret_cast<const v4f*>(base + (size_t)row * 512u) + qw;
        v4f v = __builtin_nontemporal_load(p4);
        sum += v.x + v.z;                      // even w only
    }

    __shared__ float sdata[256];
    sdata[tid] = sum;
    __syncthreads();
#pragma unroll
    for (int s = 128; s > 0; s >>= 1) {
        if (tid < s) sdata[tid] += sdata[tid + s];
        __syncthreads();
    }
    if (tid == 0) pooled[plane] = sdata[0] * (1.0f / 65536.0f);
}

// ---------------------------------------------------------------------------
// Kernel 2: fused MLP via V_WMMA_F32_16X16X4_F32, one wave (32 threads).
//   hidden = relu(pooled @ fc1_w^T)   [8,32]  (padded to M=16)
//   out    = hidden @ fc2_w^T          [8,128]
// A-frag (16x4 f32, 2 VGPR): lane L, m=L%16, kb=2*(L/16):  a[v]=A[m][k+kb+v]
// B-frag (4x16 f32, 2 VGPR): lane L, n=L%16, kb=2*(L/16):  b[v]=B[k+kb+v][n]
// C/D    (16x16 f32, 8 VGPR): lane L, n=L%16:              c[v]=C[v+8*(L/16)][n]
// ---------------------------------------------------------------------------
__global__ __launch_bounds__(32) void mlp_wmma_kernel(
    const float* __restrict__ pooled,  // [8,128]
    const float* __restrict__ fc1,     // [32,128]
    const float* __restrict__ fc2,     // [128,32]
    float* __restrict__ out)           // [8,128]
{
    __shared__ float Apad[16][128];    // pooled zero-padded to M=16
    __shared__ float Hid[16][32];      // hidden, padded to M=16

    const int lane = threadIdx.x;      // 0..31 (one full wave; EXEC all 1s)
    const int lm   = lane & 15;        // m (or n) index within tile
    const int kb   = (lane >> 4) * 2;  // K sub-offset per lane-half

    // Stage pooled into LDS, zero rows 8..15.
    for (int idx = lane; idx < 16 * 128; idx += 32) {
        const int r = idx >> 7, c = idx & 127;
        Apad[r][c] = (r < 8) ? pooled[r * 128 + c] : 0.0f;
    }
    __syncthreads();

    // ---- GEMM1: [16x128] x [128x32] -> [16x32], K in steps of 4 ----
    v8f acc0 = {}, acc1 = {};
#pragma unroll 4
    for (int k = 0; k < 128; k += 4) {
        v2f a;
        a.x = Apad[lm][k + kb + 0];
        a.y = Apad[lm][k + kb + 1];
        // N-tile 0: n = lm
        v2f b0;
        b0.x = fc1[lm * 128 + k + kb + 0];       // B[k][n] = fc1_w[n][k]
        b0.y = fc1[lm * 128 + k + kb + 1];
        acc0 = __builtin_amdgcn_wmma_f32_16x16x4_f32(
            false, a, false, b0, (short)0, acc0, false, false);
        // N-tile 1: n = 16 + lm
        v2f b1;
        b1.x = fc1[(16 + lm) * 128 + k + kb + 0];
        b1.y = fc1[(16 + lm) * 128 + k + kb + 1];
        acc1 = __builtin_amdgcn_wmma_f32_16x16x4_f32(
            false, a, false, b1, (short)0, acc1, false, false);
    }

    // ReLU + spill hidden to LDS (C/D layout -> row-major).
    const int mrow = (lane >> 4) * 8;
#pragma unroll
    for (int v = 0; v < 8; ++v) {
        Hid[mrow + v][lm]      = fmaxf(acc0[v], 0.0f);
        Hid[mrow + v][16 + lm] = fmaxf(acc1[v], 0.0f);
    }
    __syncthreads();

    // ---- GEMM2: [16x32] x [32x128] -> [16x128] ----
#pragma unroll
    for (int nt = 0; nt < 8; ++nt) {
        v8f c = {};
#pragma unroll
        for (int k = 0; k < 32; k += 4) {
            v2f a;
            a.x = Hid[lm][k + kb + 0];
            a.y = Hid[lm][k + kb + 1];
            const int n = nt * 16 + lm;
            v2f b;
            b.x = fc2[n * 32 + k + kb + 0];      // B[k][n] = fc2_w[n][k]
            b.y = fc2[n * 32 + k + kb + 1];
            c = __builtin_amdgcn_wmma_f32_16x16x4_f32(
                false, a, false, b, (short)0, c, false, false);
        }
        // Write valid rows m<8 of the D tile: out[m][nt*16 + lm]
#pragma unroll
        for (int v = 0; v < 8; ++v) {
            const int m = mrow + v;
            if (m < 8) out[m * 128 + nt * 16 + lm] = c[v];
        }
    }
}

// ---------------------------------------------------------------------------
// Launch: reduction feeds pooled through d_ws, then single-wave WMMA MLP.
// Input order: a, b, attn_w, attn_b, fc1_w, fc2_w. (a/attn_* are dead.)
// ---------------------------------------------------------------------------
extern "C" void kernel_launch(void* const* d_in, const int* in_sizes, int n_in,
                              void* d_out, int out_size, void* d_ws, size_t ws_size,
                              hipStream_t stream) {
    const float* bsrc = (const float*)d_in[1];
    const float* fc1  = (const float*)d_in[4];
    const float* fc2  = (const float*)d_in[5];
    float* pooled = (float*)d_ws;          // 1024 floats scratch
    float* out    = (float*)d_out;         // 1024 floats

    pool_even_kernel<<<1024, 256, 0, stream>>>(bsrc, pooled);
    mlp_wmma_kernel<<<1, 32, 0, stream>>>(pooled, fc1, fc2, out);
}